// DeltaNet_2551210573874
// MI455X (gfx1250) — compile-verified
//
#include <hip/hip_runtime.h>

// ---------------------------------------------------------------------------
// DeltaNet forward for MI455X (gfx1250, wave32, WMMA).
// All math in fp32 using V_WMMA_F32_16X16X4_F32 so results match the fp32
// reference bit-for-bit up to reduction order.
// Workspace requirement: ~340 MB (5x (B*L*P) f32 buffers + beta + state).
// ---------------------------------------------------------------------------

#define HID 1024
#define NH  4
#define DH  256          // head dim
#define PD  1024         // proj dim = NH*DH
#define KW  4
#define CHK 64           // chunk size
#define LSEQ 4096
#define BATCH 4
#define BL (BATCH * LSEQ)        // 16384 rows
#define NCHUNK (LSEQ / CHK)      // 64

typedef __attribute__((ext_vector_type(2))) float v2f;
typedef __attribute__((ext_vector_type(8))) float v8f;

__device__ __forceinline__ v8f wmma4(v2f a, v2f b, v8f c) {
  // D = A(16x4,f32) * B(4x16,f32) + C(16x16,f32)
  return __builtin_amdgcn_wmma_f32_16x16x4_f32(false, a, false, b, (short)0, c,
                                               false, false);
}
__device__ __forceinline__ v8f zero8() { return (v8f)0.0f; }

// ---------------------------------------------------------------------------
// Generic fp32 WMMA GEMM: C[M,N] = A[M,K] @ B[K,N]   (M%128==0, N%64==0, K%16==0)
// Block: 256 threads = 8 waves in a 4x2 grid, each wave owns a 32x32 tile.
// ---------------------------------------------------------------------------
__global__ __launch_bounds__(256) void gemm_f32_wmma(
    const float* __restrict__ A, const float* __restrict__ B,
    float* __restrict__ C, int M, int N, int K) {
  __shared__ float As[128][20];   // +4 pad keeps rows 16B aligned, rotates banks
  __shared__ float Bs[16][68];

  const int tid  = threadIdx.x;
  const int w    = tid >> 5;
  const int lane = tid & 31;
  const int lo   = lane & 15;
  const int half = lane >> 4;
  const int wr   = w >> 1;        // 0..3
  const int wc   = w & 1;         // 0..1
  const int m0   = blockIdx.y * 128;
  const int n0   = blockIdx.x * 64;

  v8f acc[2][2];
  acc[0][0] = zero8(); acc[0][1] = zero8();
  acc[1][0] = zero8(); acc[1][1] = zero8();

  for (int kb = 0; kb < K; kb += 16) {
    // stage A panel 128x16 (2 float4 per thread)
#pragma unroll
    for (int it = 0; it < 2; ++it) {
      int fid = tid + it * 256;
      int r = fid >> 2, cq = (fid & 3) << 2;
      *(float4*)&As[r][cq] =
          *(const float4*)(A + (size_t)(m0 + r) * K + kb + cq);
    }
    // stage B panel 16x64 (1 float4 per thread)
    {
      int r = tid >> 4, cq = (tid & 15) << 2;
      *(float4*)&Bs[r][cq] =
          *(const float4*)(B + (size_t)(kb + r) * N + n0 + cq);
    }
    if (kb + 16 < K) {  // hint next A panel into cache (global_prefetch_b8)
      __builtin_prefetch(A + (size_t)(m0 + (tid >> 1)) * K + kb + 16, 0, 1);
    }
    __syncthreads();

#pragma unroll
    for (int kk = 0; kk < 16; kk += 4) {
      const int k2 = kk + half * 2;
      v2f a0, a1, b0, b1;
      a0.x = As[wr * 32 + lo][k2];      a0.y = As[wr * 32 + lo][k2 + 1];
      a1.x = As[wr * 32 + 16 + lo][k2]; a1.y = As[wr * 32 + 16 + lo][k2 + 1];
      b0.x = Bs[k2][wc * 32 + lo];      b0.y = Bs[k2 + 1][wc * 32 + lo];
      b1.x = Bs[k2][wc * 32 + 16 + lo]; b1.y = Bs[k2 + 1][wc * 32 + 16 + lo];
      acc[0][0] = wmma4(a0, b0, acc[0][0]);
      acc[0][1] = wmma4(a0, b1, acc[0][1]);
      acc[1][0] = wmma4(a1, b0, acc[1][0]);
      acc[1][1] = wmma4(a1, b1, acc[1][1]);
    }
    __syncthreads();
  }

#pragma unroll
  for (int i = 0; i < 2; ++i)
#pragma unroll
    for (int j = 0; j < 2; ++j)
#pragma unroll
      for (int v = 0; v < 8; ++v) {
        int m = m0 + wr * 32 + i * 16 + v + 8 * half;
        int n = n0 + wc * 32 + j * 16 + lo;
        C[(size_t)m * N + n] = acc[i][j][v];
      }
}

// ---------------------------------------------------------------------------
// Depthwise causal conv (KW=4) + SiLU (+ optional per-head L2 norm).
// mode: 0 = none (v), 1 = l2 norm (k), 2 = l2 norm * D^-1/2 (q)
// grid = B*L blocks, 256 threads; thread t handles channel h*256+t per head.
// ---------------------------------------------------------------------------
__global__ __launch_bounds__(256) void conv_silu_norm_kernel(
    const float* __restrict__ pre, const float* __restrict__ wconv,
    float* __restrict__ out, int mode) {
  __shared__ float red[256];
  const int tid = threadIdx.x;
  const int bl  = blockIdx.x;
  const int l   = bl & (LSEQ - 1);

  for (int h = 0; h < NH; ++h) {
    const int p = h * DH + tid;
    float s = 0.f;
#pragma unroll
    for (int j = 0; j < KW; ++j) {
      int ll = l - (KW - 1) + j;
      if (ll >= 0)
        s += pre[(size_t)(bl - l + ll) * PD + p] * wconv[p * KW + j];
    }
    float val = s / (1.f + expf(-s));  // SiLU
    if (mode != 0) {
      red[tid] = val * val;
      __syncthreads();
      for (int off = 128; off > 0; off >>= 1) {
        if (tid < off) red[tid] += red[tid + off];
        __syncthreads();
      }
      float nrm = red[0];
      __syncthreads();
      float rn = rsqrtf(fmaxf(nrm, 1e-24f));
      if (mode == 2) rn *= 0.0625f;  // D^-1/2, D=256
      val *= rn;
    }
    out[(size_t)bl * PD + p] = val;
  }
}

// ---------------------------------------------------------------------------
// beta = sigmoid(x @ Wb)   (B*L, H)
// ---------------------------------------------------------------------------
__global__ __launch_bounds__(256) void beta_kernel(
    const float* __restrict__ x, const float* __restrict__ Wb,
    float* __restrict__ beta) {
  const int i  = blockIdx.x * 256 + threadIdx.x;  // < BL*NH
  const int bl = i >> 2;
  const int h  = i & 3;
  const float* xr = x + (size_t)bl * HID;
  float s = 0.f;
  for (int k = 0; k < HID; k += 4) {
    float4 xv = *(const float4*)(xr + k);
    s += xv.x * Wb[(k + 0) * NH + h] + xv.y * Wb[(k + 1) * NH + h] +
         xv.z * Wb[(k + 2) * NH + h] + xv.w * Wb[(k + 3) * NH + h];
  }
  beta[i] = 1.f / (1.f + expf(-s));
}

// ---------------------------------------------------------------------------
// RMSNorm over head dim (in place), per (b,l,h).
// ---------------------------------------------------------------------------
__global__ __launch_bounds__(256) void rmsnorm_kernel(
    float* __restrict__ o, const float* __restrict__ rw) {
  __shared__ float red[256];
  const int tid = threadIdx.x;
  const int bl  = blockIdx.x;
  for (int h = 0; h < NH; ++h) {
    const size_t idx = (size_t)bl * PD + h * DH + tid;
    float v = o[idx];
    red[tid] = v * v;
    __syncthreads();
    for (int off = 128; off > 0; off >>= 1) {
      if (tid < off) red[tid] += red[tid + off];
      __syncthreads();
    }
    float mean = red[0] * (1.f / DH);
    __syncthreads();
    o[idx] = v * rsqrtf(mean + 1e-5f) * rw[tid];
  }
}

// ---------------------------------------------------------------------------
// Chunked delta-rule scan. One block (512 thr = 16 wave32) per (b,h).
// State S kept transposed in global workspace: Sg[e][d] = S[d][e]  (L2 resident,
// 256KB per (b,h)).  Chunk working set lives in ~295KB of dynamic LDS.
// ---------------------------------------------------------------------------
__global__ __launch_bounds__(512) void scan_kernel(
    const float* __restrict__ qp, const float* __restrict__ kp,
    const float* __restrict__ vp, const float* __restrict__ betap,
    float* __restrict__ Sg_all, float* __restrict__ oat,
    float* __restrict__ Sout) {
  extern __shared__ float smem[];
  float* Qs  = smem;             // 64 x 260  (Q, pre-scaled by D^-1/2)
  float* Ks  = Qs + 64 * 260;    // 64 x 260
  float* Ws  = Ks + 64 * 260;    // 64 x 260  (W = T@Kb)
  float* Us  = Ws + 64 * 260;    // 64 x 260  (U, then mid)
  float* ATm = Us + 64 * 260;    // 64 x 68   (A, then T in place)
  float* ATT = ATm + 64 * 68;    // 64 x 68   (tril(Q K^T))
  float* bts = ATT + 64 * 68;    // 64
  float* rwb = bts + 64;         // 64

  const int tid  = threadIdx.x;
  const int w    = tid >> 5;
  const int lane = tid & 31;
  const int lo   = lane & 15;
  const int half = lane >> 4;
  const int bh   = blockIdx.x;
  const int b    = bh >> 2;
  const int h    = bh & 3;

  float* Sg = Sg_all + (size_t)bh * (DH * DH);            // [e][d]
  const size_t baseRow = (size_t)(b * LSEQ) * PD + (size_t)h * DH;

  for (int i = tid; i < DH * DH; i += 512) Sg[i] = 0.f;
  __threadfence();
  __syncthreads();

  for (int ci = 0; ci < NCHUNK; ++ci) {
    const int l0 = ci * CHK;

    // ---- load Q,K chunk + beta into LDS ----
    {
      const float* qb = qp + baseRow + (size_t)l0 * PD;
      const float* kb = kp + baseRow + (size_t)l0 * PD;
      for (int f = tid; f < 64 * 64; f += 512) {           // float4 units
        int c = f >> 6, q4 = (f & 63) << 2;
        *(float4*)&Qs[c * 260 + q4] = *(const float4*)(qb + (size_t)c * PD + q4);
        *(float4*)&Ks[c * 260 + q4] = *(const float4*)(kb + (size_t)c * PD + q4);
      }
      if (tid < 64) bts[tid] = betap[(size_t)(b * LSEQ + l0 + tid) * NH + h];
    }
    __syncthreads();

    // ---- A = I + stril(Kb K^T) and attn = tril(Q K^T): 16 tiles / 16 waves --
    {
      const int tr = w >> 2, tc = w & 3;
      v8f accA = zero8(), accQ = zero8();
      if (tr >= tc) {
        const float* Arow = &Ks[(tr * 16) * 260];
        const float* Qrow = &Qs[(tr * 16) * 260];
        const float* Bcol = &Ks[(tc * 16) * 260];
        const float betam = bts[tr * 16 + lo];
        for (int e = 0; e < DH; e += 4) {
          const int k2 = e + half * 2;
          v2f ak, aq, bb;
          ak.x = Arow[lo * 260 + k2] * betam;
          ak.y = Arow[lo * 260 + k2 + 1] * betam;
          aq.x = Qrow[lo * 260 + k2];
          aq.y = Qrow[lo * 260 + k2 + 1];
          bb.x = Bcol[lo * 260 + k2];        // K^T: element(k,n)=K[n][k]
          bb.y = Bcol[lo * 260 + k2 + 1];
          accA = wmma4(ak, bb, accA);
          accQ = wmma4(aq, bb, accQ);
        }
      }
#pragma unroll
      for (int v = 0; v < 8; ++v) {
        int c1 = tr * 16 + v + 8 * half, c2 = tc * 16 + lo;
        ATm[c1 * 68 + c2] = (c1 > c2) ? accA[v] : (c1 == c2 ? 1.f : 0.f);
        ATT[c1 * 68 + c2] = (c1 >= c2) ? accQ[v] : 0.f;
      }
    }
    __syncthreads();

    // ---- T = A^{-1} (unit lower triangular), in place, forward subst. ----
    for (int i = 1; i < 64; ++i) {
      if (tid < 64) rwb[tid] = ATm[i * 68 + tid];  // original row i of A
      __syncthreads();
      if (tid < i) {
        float s = rwb[tid];
        for (int k = tid + 1; k < i; ++k) s += rwb[k] * ATm[k * 68 + tid];
        ATm[i * 68 + tid] = -s;
      }
      __syncthreads();
    }

    // ---- W = T @ Kb, U = T @ Vb  (V streamed from global) ----
    {
      const int tr = w >> 2, dg = (w & 3) * 4;
      v8f accW[4], accU[4];
#pragma unroll
      for (int j = 0; j < 4; ++j) { accW[j] = zero8(); accU[j] = zero8(); }
      const float* Trow  = &ATm[(tr * 16) * 68];
      const float* vbase = vp + baseRow + (size_t)l0 * PD;
      for (int kk = 0; kk < 64; kk += 4) {
        const int k2 = kk + half * 2;
        v2f at;
        at.x = Trow[lo * 68 + k2];
        at.y = Trow[lo * 68 + k2 + 1];
        const float b0 = bts[k2], b1 = bts[k2 + 1];
#pragma unroll
        for (int j = 0; j < 4; ++j) {
          const int d0 = (dg + j) * 16;
          v2f bk, bv;
          bk.x = Ks[k2 * 260 + d0 + lo] * b0;
          bk.y = Ks[(k2 + 1) * 260 + d0 + lo] * b1;
          bv.x = vbase[(size_t)k2 * PD + d0 + lo] * b0;
          bv.y = vbase[(size_t)(k2 + 1) * PD + d0 + lo] * b1;
          accW[j] = wmma4(at, bk, accW[j]);
          accU[j] = wmma4(at, bv, accU[j]);
        }
      }
#pragma unroll
      for (int j = 0; j < 4; ++j) {
        const int d0 = (dg + j) * 16;
#pragma unroll
        for (int v = 0; v < 8; ++v) {
          const int m = tr * 16 + v + 8 * half;
          Ws[m * 260 + d0 + lo] = accW[j][v];
          Us[m * 260 + d0 + lo] = accU[j][v];
        }
      }
    }
    __syncthreads();

    // ---- fused: accM = W@S^T, accO = Q@S^T ; mid = U - accM (into Us) ----
    v8f accO[4];
    {
      const int tr = w >> 2, dg = (w & 3) * 4;
      v8f accM[4];
#pragma unroll
      for (int j = 0; j < 4; ++j) { accM[j] = zero8(); accO[j] = zero8(); }
      const float* Wrow = &Ws[(tr * 16) * 260];
      const float* Qrow = &Qs[(tr * 16) * 260];
      for (int e = 0; e < DH; e += 4) {
        const int k2 = e + half * 2;
        v2f aw, aq;
        aw.x = Wrow[lo * 260 + k2]; aw.y = Wrow[lo * 260 + k2 + 1];
        aq.x = Qrow[lo * 260 + k2]; aq.y = Qrow[lo * 260 + k2 + 1];
#pragma unroll
        for (int j = 0; j < 4; ++j) {
          const int d0 = (dg + j) * 16;
          v2f bs;                                   // St element (e,d)=Sg[e][d]
          bs.x = Sg[(size_t)k2 * DH + d0 + lo];
          bs.y = Sg[(size_t)(k2 + 1) * DH + d0 + lo];
          accM[j] = wmma4(aw, bs, accM[j]);
          accO[j] = wmma4(aq, bs, accO[j]);
        }
      }
#pragma unroll
      for (int j = 0; j < 4; ++j) {
        const int d0 = (dg + j) * 16;
#pragma unroll
        for (int v = 0; v < 8; ++v) {
          float* up = &Us[(tr * 16 + v + 8 * half) * 260 + d0 + lo];
          *up = *up - accM[j][v];                   // mid = U - W@S^T
        }
      }
    }
    __syncthreads();

    // ---- O = Q@S^T + attn @ mid ; store ----
    {
      const int tr = w >> 2, dg = (w & 3) * 4;
      const float* Arow = &ATT[(tr * 16) * 68];
      for (int kk = 0; kk < 64; kk += 4) {
        const int k2 = kk + half * 2;
        v2f aa;
        aa.x = Arow[lo * 68 + k2];
        aa.y = Arow[lo * 68 + k2 + 1];
#pragma unroll
        for (int j = 0; j < 4; ++j) {
          const int d0 = (dg + j) * 16;
          v2f bm;
          bm.x = Us[k2 * 260 + d0 + lo];
          bm.y = Us[(k2 + 1) * 260 + d0 + lo];
          accO[j] = wmma4(aa, bm, accO[j]);
        }
      }
      float* ob = oat + baseRow + (size_t)l0 * PD;
#pragma unroll
      for (int j = 0; j < 4; ++j) {
        const int d0 = (dg + j) * 16;
#pragma unroll
        for (int v = 0; v < 8; ++v)
          ob[(size_t)(tr * 16 + v + 8 * half) * PD + d0 + lo] = accO[j][v];
      }
    }
    __syncthreads();

    // ---- state update: Sg[e][d] += K^T @ mid  (16 e-tiles / 16 waves) ----
    {
      const int et = w;
      for (int dt = 0; dt < 16; ++dt) {
        const int d0 = dt * 16;
        v8f acc;
#pragma unroll
        for (int v = 0; v < 8; ++v)
          acc[v] = Sg[(size_t)(et * 16 + v + 8 * half) * DH + d0 + lo];
        for (int kk = 0; kk < 64; kk += 4) {
          const int k2 = kk + half * 2;
          v2f at, bm;                        // A^T: element(m=e,k=c)=K[c][e]
          at.x = Ks[k2 * 260 + et * 16 + lo];
          at.y = Ks[(k2 + 1) * 260 + et * 16 + lo];
          bm.x = Us[k2 * 260 + d0 + lo];
          bm.y = Us[(k2 + 1) * 260 + d0 + lo];
          acc = wmma4(at, bm, acc);
        }
#pragma unroll
        for (int v = 0; v < 8; ++v)
          Sg[(size_t)(et * 16 + v + 8 * half) * DH + d0 + lo] = acc[v];
      }
    }
    __threadfence();
    __syncthreads();
  }

  // final state: Sout (B,H,D,D) with S[d][e] = Sg[e][d]
  for (int i = tid; i < DH * DH; i += 512) {
    const int d = i >> 8, e = i & 255;
    Sout[(size_t)bh * DH * DH + i] = Sg[(size_t)e * DH + d];
  }
}

// ---------------------------------------------------------------------------
// host launcher
// ---------------------------------------------------------------------------
extern "C" void kernel_launch(void* const* d_in, const int* in_sizes, int n_in,
                              void* d_out, int out_size, void* d_ws,
                              size_t ws_size, hipStream_t stream) {
  const float* x    = (const float*)d_in[0];
  const float* Wq   = (const float*)d_in[1];
  const float* Wk   = (const float*)d_in[2];
  const float* Wv   = (const float*)d_in[3];
  const float* Wb   = (const float*)d_in[4];
  const float* cqw  = (const float*)d_in[5];
  const float* ckw  = (const float*)d_in[6];
  const float* cvw  = (const float*)d_in[7];
  const float* rmsw = (const float*)d_in[8];
  const float* Wo   = (const float*)d_in[9];
  float* out = (float*)d_out;

  // workspace partition (floats); needs ~340MB
  float* ws    = (float*)d_ws;
  float* pre   = ws;                         // BL*PD (reused q->k->v)
  float* qpost = pre   + (size_t)BL * PD;
  float* kpost = qpost + (size_t)BL * PD;
  float* vpost = kpost + (size_t)BL * PD;
  float* betab = vpost + (size_t)BL * PD;    // BL*NH
  float* Sg    = betab + (size_t)BL * NH;    // B*H*D*D
  float* oat   = Sg + (size_t)BATCH * NH * DH * DH;  // BL*PD

  const dim3 gproj(PD / 64, BL / 128);       // (16, 128)
  const size_t scan_lds =
      (size_t)(4 * 64 * 260 + 2 * 64 * 68 + 128) * sizeof(float);  // ~295KB
  (void)hipFuncSetAttribute((const void*)scan_kernel,
                            hipFuncAttributeMaxDynamicSharedMemorySize,
                            (int)scan_lds);

  // q/k/v projections + causal conv + SiLU + norm
  gemm_f32_wmma<<<gproj, 256, 0, stream>>>(x, Wq, pre, BL, PD, HID);
  conv_silu_norm_kernel<<<BL, 256, 0, stream>>>(pre, cqw, qpost, 2);
  gemm_f32_wmma<<<gproj, 256, 0, stream>>>(x, Wk, pre, BL, PD, HID);
  conv_silu_norm_kernel<<<BL, 256, 0, stream>>>(pre, ckw, kpost, 1);
  gemm_f32_wmma<<<gproj, 256, 0, stream>>>(x, Wv, pre, BL, PD, HID);
  conv_silu_norm_kernel<<<BL, 256, 0, stream>>>(pre, cvw, vpost, 0);
  beta_kernel<<<(BL * NH) / 256, 256, 0, stream>>>(x, Wb, betab);

  // chunked delta-rule scan; writes attention output + final state
  scan_kernel<<<BATCH * NH, 512, scan_lds, stream>>>(
      qpost, kpost, vpost, betab, Sg, oat, out + (size_t)BL * HID);

  // RMSNorm + output projection
  rmsnorm_kernel<<<BL, 256, 0, stream>>>(oat, rmsw);
  gemm_f32_wmma<<<dim3(HID / 64, BL / 128), 256, 0, stream>>>(
      oat, Wo, out, BL, HID, PD);
}